// interactionModule_30459908063878
// MI455X (gfx1250) — compile-verified
//
#include <hip/hip_runtime.h>
#include <math.h>

// ---------------------------------------------------------------------------
// Reference constants
// ---------------------------------------------------------------------------
#define GAMMA  0.1f
#define MIN_R  0.1f
#define EPS_N  1e-12f

typedef long long i64;

#define CHUNK 128                 // edges per wave per pipeline stage
#define WAVES_PER_BLOCK 8         // 256 threads, wave32

// ---------------------------------------------------------------------------
// CDNA5 async global->LDS copy (ASYNCcnt-tracked DMA into LDS).
//   lptr low 32 bits == workgroup-relative LDS byte address (ISA 10.2).
// Emits: global_load_async_to_lds_b128 vDst, v[a:a+1], off
// ---------------------------------------------------------------------------
__device__ __forceinline__ void async_copy_b128(const void* gptr, void* lptr) {
    unsigned           lds  = (unsigned)(unsigned long long)lptr;
    unsigned long long addr = (unsigned long long)gptr;
    asm volatile("global_load_async_to_lds_b128 %0, %1, off"
                 :
                 : "v"(lds), "v"(addr));
}

__device__ __forceinline__ void wait_async_le4() {
    asm volatile("s_wait_asynccnt 4" ::: "memory");
}
__device__ __forceinline__ void wait_async_0() {
    asm volatile("s_wait_asynccnt 0" ::: "memory");
}

// ---------------------------------------------------------------------------
// Per-edge LJ message + scatter (atomic, no-return -> STOREcnt path)
//   r    = max(|dr|, MIN_R);  s = 1/r
//   f    = 4*s^6*(12*s^6 - 6)/r
//   msg  = f * dr / max(|dr|, 1e-12)
// ---------------------------------------------------------------------------
__device__ __forceinline__ void edge_compute(const float* __restrict__ x,
                                             int s, int d,
                                             float* __restrict__ out) {
    float dx = x[3 * d + 0] - x[3 * s + 0];
    float dy = x[3 * d + 1] - x[3 * s + 1];
    float dz = x[3 * d + 2] - x[3 * s + 2];
    float r2 = dx * dx + dy * dy + dz * dz;
    float ab = sqrtf(r2);
    float invn = 1.0f / fmaxf(ab, EPS_N);        // normalize denominator
    float r  = fmaxf(ab, MIN_R);                 // clamped radius for force
    float ir = 1.0f / r;
    float s2 = ir * ir;
    float s6 = s2 * s2 * s2;
    float f  = s6 * (48.0f * s6 - 24.0f) * ir;   // == 4*s^6*(12*s^6-6)/r
    float c  = f * invn;
    atomicAdd(&out[3 * d + 0], c * dx);
    atomicAdd(&out[3 * d + 1], c * dy);
    atomicAdd(&out[3 * d + 2], c * dz);
}

// ---------------------------------------------------------------------------
// out = -GAMMA * v   (also clears the 0xAA poison in d_out)
// ---------------------------------------------------------------------------
__global__ void init_out(const float* __restrict__ v, float* __restrict__ out,
                         int n) {
    int i = blockIdx.x * blockDim.x + threadIdx.x;
    if (i < n) out[i] = -GAMMA * v[i];
}

// ---------------------------------------------------------------------------
// Main edge kernel: per-wave double-buffered async staging of the int64
// src/dst index streams into LDS, compute from LDS (ds_load_b32 of the low
// dword of each index).  Each wave owns chunks gw, gw+W, gw+2W, ...
// ---------------------------------------------------------------------------
__global__ __launch_bounds__(256) void edge_async(
    const float* __restrict__ x,
    const i64*   __restrict__ src,
    const i64*   __restrict__ dst,
    float*       __restrict__ out,
    int numChunks, int totalWaves) {
    // [buffer][wave][src/dst][edge]  = 2*8*2*128*8B = 32 KB
    __shared__ i64 sm[2][WAVES_PER_BLOCK][2][CHUNK];

    const int lane = threadIdx.x & 31;
    const int wib  = threadIdx.x >> 5;
    const int gw   = blockIdx.x * WAVES_PER_BLOCK + wib;
    if (gw >= numChunks) return;   // uniform per wave

    // Stage one CHUNK of src+dst indices: 4 x b128 per lane
    // (each lane moves 2 int64 per instruction; 32 lanes cover 64 elements).
    auto stage = [&](int c, int bb) {
        const i64* gs = src + (size_t)c * CHUNK + lane * 2;
        const i64* gd = dst + (size_t)c * CHUNK + lane * 2;
        async_copy_b128(gs,      &sm[bb][wib][0][lane * 2]);
        async_copy_b128(gs + 64, &sm[bb][wib][0][lane * 2 + 64]);
        async_copy_b128(gd,      &sm[bb][wib][1][lane * 2]);
        async_copy_b128(gd + 64, &sm[bb][wib][1][lane * 2 + 64]);
    };

    stage(gw, 0);
    int b = 0;
    for (int c = gw; c < numChunks; c += totalWaves) {
        const int  cn   = c + totalWaves;
        const bool more = (cn < numChunks);
        if (more) {
            stage(cn, b ^ 1);      // prefetch next chunk while we compute
            wait_async_le4();      // current buffer's 4 copies are done
        } else {
            wait_async_0();
        }
        // 32-bit views of this wave's staged index buffers (low dword of
        // each little-endian int64; indices are < 2^31).
        const int* sp = (const int*)&sm[b][wib][0][0];
        const int* dp = (const int*)&sm[b][wib][1][0];
#pragma unroll
        for (int j = 0; j < 4; ++j) {
            int i = lane + 32 * j;
            int s = sp[2 * i];
            int d = dp[2 * i];
            edge_compute(x, s, d, out);
        }
        b ^= 1;
    }
}

// ---------------------------------------------------------------------------
// Tail: edges not covered by full CHUNKs
// ---------------------------------------------------------------------------
__global__ void edge_tail(const float* __restrict__ x,
                          const i64*   __restrict__ src,
                          const i64*   __restrict__ dst,
                          float*       __restrict__ out,
                          int base, int E) {
    int e = base + blockIdx.x * blockDim.x + threadIdx.x;
    if (e < E) edge_compute(x, (int)src[e], (int)dst[e], out);
}

// ---------------------------------------------------------------------------
// Host entry
//   d_in: [x (N*3 f32), v (N*3 f32), src (E i64), dst (E i64), n_nodes]
//   d_out: N*3 f32
// ---------------------------------------------------------------------------
extern "C" void kernel_launch(void* const* d_in, const int* in_sizes, int n_in,
                              void* d_out, int out_size, void* d_ws,
                              size_t ws_size, hipStream_t stream) {
    const float* x   = (const float*)d_in[0];
    const float* v   = (const float*)d_in[1];
    const i64*   src = (const i64*)d_in[2];
    const i64*   dst = (const i64*)d_in[3];
    float*       out = (float*)d_out;

    const int nOut = out_size;       // = 3 * n_nodes
    const int E    = in_sizes[2];    // edge count

    init_out<<<(nOut + 255) / 256, 256, 0, stream>>>(v, out, nOut);

    const int numChunks = E / CHUNK;
    const int tailBase  = numChunks * CHUNK;
    const int tail      = E - tailBase;

    if (numChunks > 0) {
        // ~4 chunks per wave so the double buffer gets reused, while still
        // launching thousands of waves to fill the machine.
        int totalWaves = (numChunks + 3) / 4;
        if (totalWaves < 1) totalWaves = 1;
        int blocks = (totalWaves + WAVES_PER_BLOCK - 1) / WAVES_PER_BLOCK;
        totalWaves = blocks * WAVES_PER_BLOCK;   // actual stride
        edge_async<<<blocks, 256, 0, stream>>>(x, src, dst, out, numChunks,
                                               totalWaves);
    }
    if (tail > 0) {
        edge_tail<<<(tail + 255) / 256, 256, 0, stream>>>(x, src, dst, out,
                                                          tailBase, E);
    }
}